// YOLOLoss_20212116095641
// MI455X (gfx1250) — compile-verified
//
#include <hip/hip_runtime.h>
#include <cstdint>
#include <cstddef>

#define NBLOCKS  512
#define NTHREADS 256
#define NTGT     512
#define NCLS     8

typedef __attribute__((ext_vector_type(2))) float v2f;
typedef __attribute__((ext_vector_type(8))) float v8f;

// min(softplus(x), 100) == -clip(log(1 - sigmoid(x)), -100)
__device__ __forceinline__ float softplus_c(float x) {
    float r = fmaxf(x, 0.0f) + log1pf(__expf(-fabsf(x)));
    return fminf(r, 100.0f);
}

// Wave32 sum-reduction using V_WMMA_F32_16X16X4_F32.
// A = ones(16x4), B vgpr0 = v (rows K0/K2 across lane halves), B vgpr1 = 0.
// D[m][n] = sum_k B[k][n]; the 16 column sums partition the 32 lane values,
// so summing d[0] over lanes 0..15 yields the full 32-lane total (in lane 0).
__device__ __forceinline__ float wave_sum_wmma(float v) {
    v2f a; a[0] = 1.0f; a[1] = 1.0f;
    v2f b; b[0] = v;    b[1] = 0.0f;
    v8f c = {0.f,0.f,0.f,0.f,0.f,0.f,0.f,0.f};
    v8f d = __builtin_amdgcn_wmma_f32_16x16x4_f32(false, a, false, b,
                                                  (short)0, c, false, false);
    float r = d[0];
    r += __shfl_xor(r, 8, 32);
    r += __shfl_xor(r, 4, 32);
    r += __shfl_xor(r, 2, 32);
    r += __shfl_xor(r, 1, 32);
    return r; // valid in lane 0
}

// ---------------------------------------------------------------------------
// Kernel 1: dense sum of min(softplus(conf_logit),100) over every cell of all
// three scales (the "noobj" BCE numerator before subtracting obj cells).
// Conf channel of anchor a is channel a*13+4 of the 39-channel tensor.
// Uses CDNA5 async global->LDS staging (ASYNCcnt path).
// ---------------------------------------------------------------------------
__global__ __launch_bounds__(NTHREADS) void yolo_dense(
        const float* __restrict__ p0, const float* __restrict__ p1,
        const float* __restrict__ p2, float* __restrict__ partials) {
    __shared__ float stage[NTHREADS * 4];
    __shared__ float wsum[NTHREADS / 32][3];

    const int   HW[3]   = {6400, 1600, 400};
    const float* preds[3] = {p0, p1, p2};
    const long long Q0 = 64LL * 3 * 6400 / 4;   // quads (float4) per scale
    const long long Q1 = 64LL * 3 * 1600 / 4;
    const long long Q2 = 64LL * 3 * 400  / 4;
    const long long QT = Q0 + Q1 + Q2;

    const int tid = threadIdx.x;
    const unsigned lds_off = (unsigned)(uintptr_t)&stage[tid * 4];

    float acc[3] = {0.f, 0.f, 0.f};

    for (long long q = (long long)blockIdx.x * NTHREADS + tid; q < QT;
         q += (long long)NBLOCKS * NTHREADS) {
        int s; long long local;
        if (q < Q0)            { s = 0; local = q; }
        else if (q < Q0 + Q1)  { s = 1; local = q - Q0; }
        else                   { s = 2; local = q - Q0 - Q1; }
        const int hw  = HW[s];
        const int qpp = hw >> 2;                 // float4 quads per plane
        const int plane = (int)(local / qpp);
        const int w4    = (int)(local - (long long)plane * qpp);
        const int b = plane / 3, a = plane - b * 3;
        const size_t off = (size_t)(b * 39 + a * 13 + 4) * hw + (size_t)w4 * 4;
        const uint64_t gaddr = (uint64_t)(uintptr_t)(preds[s] + off);

        // CDNA5 async tensor/memory path: global -> LDS, track with ASYNCcnt.
        asm volatile("global_load_async_to_lds_b128 %0, %1, off\n\t"
                     "s_wait_asynccnt 0"
                     :: "v"(lds_off), "v"(gaddr) : "memory");

        const float4 v = *(const float4*)&stage[tid * 4];
        acc[s] += softplus_c(v.x) + softplus_c(v.y) +
                  softplus_c(v.z) + softplus_c(v.w);
    }

    const int lane = tid & 31, wv = tid >> 5;
    for (int s = 0; s < 3; ++s) {
        float v = acc[s];
        for (int m = 16; m >= 1; m >>= 1) v += __shfl_xor(v, m, 32);
        if (lane == 0) wsum[wv][s] = v;
    }
    __syncthreads();
    if (tid < 3) {
        float v = 0.f;
        for (int w = 0; w < NTHREADS / 32; ++w) v += wsum[w][tid];
        partials[(size_t)blockIdx.x * 3 + tid] = v;   // deterministic
    }
}

// ---------------------------------------------------------------------------
// Kernel 2: single wave32. Target preprocessing, last-writer-wins scatter
// dedup, sparse obj losses, dense-partial reduction, final scalar.
// ---------------------------------------------------------------------------
__global__ __launch_bounds__(32) void yolo_sparse(
        const float* __restrict__ p0, const float* __restrict__ p1,
        const float* __restrict__ p2, const float* __restrict__ tgt,
        const int* __restrict__ idim_p, const float* __restrict__ partials,
        float* __restrict__ out) {
    __shared__ int    s_cell[3][NTGT];
    __shared__ float4 s_t[3][NTGT];
    __shared__ int    s_tc[NTGT];

    const int   Wd[3] = {80, 40, 20};
    const float anc[3][3][2] = {
        {{10.f, 13.f}, {16.f, 30.f}, {33.f, 23.f}},
        {{30.f, 61.f}, {62.f, 45.f}, {59.f, 119.f}},
        {{116.f, 90.f}, {156.f, 198.f}, {373.f, 326.f}}};
    const float* preds[3] = {p0, p1, p2};
    const float input_dim = (float)idim_p[0];
    const int tid = threadIdx.x;

    // ---- Phase A: per-target / per-scale precompute ----
    for (int n = tid; n < NTGT; n += 32) {
        const float* t = tgt + (size_t)n * 6;
        const int b  = (int)t[0];
        const int tc = (int)t[1];
        s_tc[n] = tc;
        for (int s = 0; s < 3; ++s) {
            const int   W  = Wd[s];
            const float fW = (float)W;                 // fH == fW (square)
            const float gx = t[2] * fW, gy = t[3] * fW;
            const float gw = t[4] * input_dim, gh = t[5] * input_dim;
            const int gi = min(max((int)gx, 0), W - 1);
            const int gj = min(max((int)gy, 0), W - 1);
            int ba = 0; float best = -1.0f;
            for (int a = 0; a < 3; ++a) {              // reference's IoU quirk:
                const float aw = anc[s][a][0] * fW;    // pixel anchor * grid size
                const float ah = anc[s][a][1] * fW;
                const float inter = fminf(gw, aw) * fminf(gh, ah);
                const float iou = inter / (gw * gh + aw * ah - inter + 1e-16f);
                if (iou > best) { best = iou; ba = a; }   // first max wins
            }
            s_cell[s][n] = ((b * 3 + ba) * W + gj) * W + gi;
            float4 tt;
            tt.x = gx - (float)gi;
            tt.y = gy - (float)gj;
            tt.z = logf(gw / anc[s][ba][0] + 1e-16f);  // scaled-factor cancels
            tt.w = logf(gh / anc[s][ba][1] + 1e-16f);
            s_t[s][n] = tt;
        }
    }
    __syncthreads();

    // ---- Phase B/C: dedup (last writer wins) + sparse obj losses ----
    float obj_acc[3] = {0.f, 0.f, 0.f};   // mse + conf-BCE + cls-BCE/8
    float sub_acc[3] = {0.f, 0.f, 0.f};   // noobj contribution to remove
    float cnt[3]     = {0.f, 0.f, 0.f};   // distinct obj cells
    for (int n = tid; n < NTGT; n += 32) {
        for (int s = 0; s < 3; ++s) {
            const int myc = s_cell[s][n];
            bool win = true;
            unsigned mask = 1u << s_tc[n];
            for (int m = 0; m < NTGT; ++m) {
                if (m == n) continue;
                if (s_cell[s][m] == myc) {
                    if (m > n) { win = false; break; }
                    mask |= 1u << s_tc[m];             // multi-hot class union
                }
            }
            if (!win) continue;
            const int W = Wd[s], hw = W * W;
            int rem = myc;
            const int gi = rem % W; rem /= W;
            const int gj = rem % W; rem /= W;
            const int a  = rem % 3;
            const int b  = rem / 3;
            const float* pp = preds[s] + (size_t)(b * 39 + a * 13) * hw +
                              (size_t)gj * W + gi;
            const float4 tt = s_t[s][n];
            const float q0 = pp[0];
            const float q1 = pp[(size_t)hw];
            const float q2 = pp[2 * (size_t)hw];
            const float q3 = pp[3 * (size_t)hw];
            const float q4 = pp[4 * (size_t)hw];
            const float px = 1.0f / (1.0f + __expf(-q0));
            const float py = 1.0f / (1.0f + __expf(-q1));
            float e = (px - tt.x) * (px - tt.x) + (py - tt.y) * (py - tt.y) +
                      (q2 - tt.z) * (q2 - tt.z) + (q3 - tt.w) * (q3 - tt.w);
            e += softplus_c(-q4);                      // obj conf BCE (t=1)
            sub_acc[s] += softplus_c(q4);              // remove from dense noobj
            float clsum = 0.f;
            for (int c = 0; c < NCLS; ++c) {
                const float z = pp[(size_t)(5 + c) * hw];
                clsum += ((mask >> c) & 1u) ? softplus_c(-z) : softplus_c(z);
            }
            obj_acc[s] += e + clsum * (1.0f / NCLS);   // cls divided by n_obj*C
            cnt[s] += 1.0f;
        }
    }

    // ---- Dense partials from kernel 1 ----
    float dsum[3] = {0.f, 0.f, 0.f};
    for (int i = tid; i < NBLOCKS; i += 32) {
        dsum[0] += partials[(size_t)i * 3 + 0];
        dsum[1] += partials[(size_t)i * 3 + 1];
        dsum[2] += partials[(size_t)i * 3 + 2];
    }

    // ---- Cross-lane reductions via WMMA (EXEC all-ones here) ----
    float objR[3], nobR[3], cntR[3];
    for (int s = 0; s < 3; ++s) {
        objR[s] = wave_sum_wmma(obj_acc[s]);
        nobR[s] = wave_sum_wmma(dsum[s] - sub_acc[s]);
        cntR[s] = wave_sum_wmma(cnt[s]);
    }

    if (tid == 0) {
        float total = 0.f;
        for (int s = 0; s < 3; ++s) {
            const float cells  = 64.0f * 3.0f * (float)(Wd[s] * Wd[s]);
            const float n_obj  = fmaxf(cntR[s], 1.0f);
            const float n_nob  = fmaxf(cells - cntR[s], 1.0f);
            total += objR[s] / n_obj + 0.5f * nobR[s] / n_nob;
        }
        out[0] = total;
    }
}

extern "C" void kernel_launch(void* const* d_in, const int* in_sizes, int n_in,
                              void* d_out, int out_size, void* d_ws, size_t ws_size,
                              hipStream_t stream) {
    const float* p0  = (const float*)d_in[0];
    const float* p1  = (const float*)d_in[1];
    const float* p2  = (const float*)d_in[2];
    const float* tgt = (const float*)d_in[3];
    const int*  idim = (const int*)d_in[4];
    float* partials  = (float*)d_ws;            // NBLOCKS*3 floats

    yolo_dense<<<NBLOCKS, NTHREADS, 0, stream>>>(p0, p1, p2, partials);
    yolo_sparse<<<1, 32, 0, stream>>>(p0, p1, p2, tgt, idim, partials,
                                      (float*)d_out);
}